// GCN_Node_35158602285142
// MI455X (gfx1250) — compile-verified
//
#include <hip/hip_runtime.h>

// ---------------------------------------------------------------------------
// 3-layer GCN on MI455X (gfx1250, wave32).
//   deg/dis -> [GEMM (fp32 WMMA 16x16x4, async-LDS-staged A tiles)
//               -> self-loop init -> edge scatter (global_atomic_add_f32)
//               -> bias(+relu)] x3
// GEMM uses V_WMMA_F32_16X16X4_F32 so numerics match the fp32 reference.
// Aggregation is the memory-bound part (~870MB gather + 870MB atomics per
// layer at 23.3 TB/s); it streams float4 (b128) gathers, one wave per edge.
// ---------------------------------------------------------------------------

typedef __attribute__((ext_vector_type(2))) float v2f;
typedef __attribute__((ext_vector_type(8))) float v8f;

// exact parameter type of __builtin_amdgcn_global_load_async_to_lds_b128
// (from clang diagnostic: "__attribute__((__vector_size__(4*sizeof(int)))) int __device__ *")
typedef int v4i_ __attribute__((vector_size(16)));
typedef __attribute__((address_space(1))) v4i_ g_v4i;   // global (AS1)
typedef __attribute__((address_space(3))) v4i_ l_v4i;   // LDS (AS3)

#define N_NODES 50000
#define N_EDGES 800000
#define KDIM    256            // inner dim of every GEMM in this model

#if __has_builtin(__builtin_amdgcn_global_load_async_to_lds_b128) && \
    __has_builtin(__builtin_amdgcn_s_wait_asynccnt)
#define HAVE_ASYNC_LDS 1
#else
#define HAVE_ASYNC_LDS 0
#endif

// ---------------- degree / normalization ----------------------------------

__global__ void deg_init_k(float* __restrict__ deg, int n) {
  int i = blockIdx.x * blockDim.x + threadIdx.x;
  if (i < n) deg[i] = 1.0f;                    // self-loop contributes 1
}

__global__ void deg_accum_k(const int* __restrict__ dst, float* __restrict__ deg, int e_cnt) {
  int e = blockIdx.x * blockDim.x + threadIdx.x;
  if (e < e_cnt) atomicAdd(&deg[dst[e]], 1.0f);
}

__global__ void dis_k(float* __restrict__ d, int n) {
  int i = blockIdx.x * blockDim.x + threadIdx.x;
  if (i < n) d[i] = rsqrtf(d[i]);              // deg >= 1 always
}

// ---------------- fp32 WMMA GEMM: C[M,N] = A[M,256] * B[256,N] -------------
// One block (8 waves) owns one 16-row M tile; the 16x256 fp32 A tile is
// staged into LDS via GLOBAL_LOAD_ASYNC_TO_LDS_B128 (ASYNCcnt) when the
// builtin exists. Row stride 260 floats -> conflict-free ds_load_b64
// fragment reads. Each wave computes DUAL ? 2 : 1 16x16 C tiles, reusing
// the same A fragment across both WMMA chains.

template <int N, bool DUAL>
__global__ __launch_bounds__(256)
void gemm_wmma_f32(const float* __restrict__ A, const float* __restrict__ B,
                   float* __restrict__ C) {
  __shared__ float As[16][260];

  const int tid  = threadIdx.x;
  const int wave = tid >> 5;
  const int lane = tid & 31;
  const int row0 = blockIdx.x << 4;

  // Stage the 16 x 256 fp32 A tile (64 float4 per row, 4 per thread).
#pragma unroll
  for (int i = tid; i < 16 * 64; i += 256) {
    const int r  = i >> 6;
    const int c4 = i & 63;
    const float4* gsrc = (const float4*)(A + (size_t)(row0 + r) * KDIM) + c4;
#if HAVE_ASYNC_LDS
    __builtin_amdgcn_global_load_async_to_lds_b128(
        (g_v4i*)(size_t)gsrc,                    // global src (AS1, int4*)
        (l_v4i*)&As[r][c4 * 4],                  // LDS dst (AS3, int4*)
        0, 0);                                   // ASYNCcnt-tracked DMA
#else
    float4 v = *gsrc;
    As[r][c4 * 4 + 0] = v.x;
    As[r][c4 * 4 + 1] = v.y;
    As[r][c4 * 4 + 2] = v.z;
    As[r][c4 * 4 + 3] = v.w;
#endif
  }
#if HAVE_ASYNC_LDS
  __builtin_amdgcn_s_wait_asynccnt(0);
#endif
  __syncthreads();

  const int m    = lane & 15;                  // A-frag row / C column-in-tile
  const int kgrp = (lane >> 4) << 1;           // lanes 16-31 take K+2,K+3
  const int col0 = wave << 4;                  // N tile 0
  const int col1 = col0 + 128;                 // N tile 1 (DUAL only)

  v8f acc0 = {};
  v8f acc1 = {};
#pragma unroll 8
  for (int k0 = 0; k0 < KDIM; k0 += 4) {
    v2f a;
    a.x = As[m][k0 + kgrp];
    a.y = As[m][k0 + kgrp + 1];
    const float* bp = B + (size_t)(k0 + kgrp) * N + m;
    v2f b0;
    b0.x = bp[col0];
    b0.y = bp[col0 + N];
    acc0 = __builtin_amdgcn_wmma_f32_16x16x4_f32(
        false, a, false, b0, (short)0, acc0, false, false);
    if (DUAL) {
      v2f b1;
      b1.x = bp[col1];
      b1.y = bp[col1 + N];
      acc1 = __builtin_amdgcn_wmma_f32_16x16x4_f32(
          false, a, false, b1, (short)0, acc1, false, false);
    }
  }

  // C layout: VGPR r holds row (r + 8*(lane>=16)), col = lane&15.
  const int crow0 = row0 + ((lane >> 4) << 3);
#pragma unroll
  for (int r = 0; r < 8; ++r) {
    C[(size_t)(crow0 + r) * N + col0 + m] = acc0[r];
    if (DUAL) C[(size_t)(crow0 + r) * N + col1 + m] = acc1[r];
  }
}

// ---------------- aggregation ----------------------------------------------

// out[i,:] = dis[i]^2 * h[i,:]   (self-loop term; fully overwrites out)
// float4-granular; csh = log2(channels).
__global__ void agg_init_k(const float* __restrict__ h, const float* __restrict__ dis,
                           float* __restrict__ out, int total4, int csh) {
  int idx4 = blockIdx.x * blockDim.x + threadIdx.x;
  if (idx4 < total4) {
    float di = dis[idx4 >> (csh - 2)];
    float s  = di * di;
    float4 v = ((const float4*)h)[idx4];
    v.x *= s; v.y *= s; v.z *= s; v.w *= s;
    ((float4*)out)[idx4] = v;
  }
}

// out[dst,:] += dis[src]*dis[dst] * h[src,:]   — one wave per edge,
// b128 gathers, dword fp32 atomics.
template <int C>
__global__ __launch_bounds__(256)
void agg_edges_k(const float* __restrict__ h, const float* __restrict__ dis,
                 const int* __restrict__ src, const int* __restrict__ dst,
                 float* __restrict__ out, int e_cnt) {
  int e = (blockIdx.x << 3) + (threadIdx.x >> 5);
  if (e >= e_cnt) return;
  int lane = threadIdx.x & 31;

  int s = src[e], d = dst[e];
  float norm = dis[s] * dis[d];
  const float4* hs = (const float4*)(h + (size_t)s * C);
  float* od = out + (size_t)d * C;

  // Hide gather latency: prefetch the next edge batch's source row.
  if (e + 8 < e_cnt) {
    int s2 = src[e + 8];
    __builtin_prefetch(h + (size_t)s2 * C + lane * 4, 0, 0);  // global_prefetch_b8
  }

#pragma unroll
  for (int c4 = lane; c4 < (C >> 2); c4 += 32) {
    float4 v = hs[c4];
    int c = c4 << 2;
    atomicAdd(&od[c + 0], norm * v.x);
    atomicAdd(&od[c + 1], norm * v.y);
    atomicAdd(&od[c + 2], norm * v.z);
    atomicAdd(&od[c + 3], norm * v.w);
  }
}

// out += bias, optional relu; float4-granular.
__global__ void bias_act_k(float* __restrict__ out, const float* __restrict__ b,
                           int total4, int csh, int do_relu) {
  int idx4 = blockIdx.x * blockDim.x + threadIdx.x;
  if (idx4 < total4) {
    int cm4 = (1 << (csh - 2)) - 1;
    float4 v  = ((float4*)out)[idx4];
    float4 bb = ((const float4*)b)[idx4 & cm4];
    v.x += bb.x; v.y += bb.y; v.z += bb.z; v.w += bb.w;
    if (do_relu) {
      v.x = fmaxf(v.x, 0.0f); v.y = fmaxf(v.y, 0.0f);
      v.z = fmaxf(v.z, 0.0f); v.w = fmaxf(v.w, 0.0f);
    }
    ((float4*)out)[idx4] = v;
  }
}

// ---------------- driver ----------------------------------------------------

extern "C" void kernel_launch(void* const* d_in, const int* in_sizes, int n_in,
                              void* d_out, int out_size, void* d_ws, size_t ws_size,
                              hipStream_t stream) {
  (void)in_sizes; (void)n_in; (void)out_size; (void)ws_size;

  const float* x   = (const float*)d_in[0];
  // d_in[1] = edge_attr (unused, matches reference)
  const int*   ei  = (const int*)d_in[2];
  const int*   src = ei;
  const int*   dst = ei + N_EDGES;
  const float* W1  = (const float*)d_in[3];
  const float* b1  = (const float*)d_in[4];
  const float* W2  = (const float*)d_in[5];
  const float* b2  = (const float*)d_in[6];
  const float* W3  = (const float*)d_in[7];
  const float* b3  = (const float*)d_in[8];
  float* out = (float*)d_out;

  float* ws   = (float*)d_ws;
  float* dis  = ws;                                  // 50000 (deg then rsqrt in place)
  float* bufA = ws + 51200;                          // 50000 x 256
  float* bufB = bufA + (size_t)N_NODES * 256;        // 50000 x 256

  const int MT   = N_NODES / 16;                     // 3125 M tiles (exact)
  const int t256 = N_NODES * 64;                     // (50000*256)/4 float4s
  const int t128 = N_NODES * 32;                     // (50000*128)/4 float4s
  const int EB   = (N_EDGES + 7) / 8;

  // normalization
  deg_init_k <<<(N_NODES + 255) / 256, 256, 0, stream>>>(dis, N_NODES);
  deg_accum_k<<<(N_EDGES + 255) / 256, 256, 0, stream>>>(dst, dis, N_EDGES);
  dis_k      <<<(N_NODES + 255) / 256, 256, 0, stream>>>(dis, N_NODES);

  // ---- layer 1: h1 = relu(Agg(x @ W1) + b1) ----
  gemm_wmma_f32<256, true><<<MT, 256, 0, stream>>>(x, W1, bufA);
  agg_init_k     <<<(t256 + 255) / 256, 256, 0, stream>>>(bufA, dis, bufB, t256, 8);
  agg_edges_k<256><<<EB, 256, 0, stream>>>(bufA, dis, src, dst, bufB, N_EDGES);
  bias_act_k     <<<(t256 + 255) / 256, 256, 0, stream>>>(bufB, b1, t256, 8, 1);

  // ---- layer 2: h2 = relu(Agg(h1 @ W2) + b2) ----
  gemm_wmma_f32<256, true><<<MT, 256, 0, stream>>>(bufB, W2, bufA);
  agg_init_k     <<<(t256 + 255) / 256, 256, 0, stream>>>(bufA, dis, bufB, t256, 8);
  agg_edges_k<256><<<EB, 256, 0, stream>>>(bufA, dis, src, dst, bufB, N_EDGES);
  bias_act_k     <<<(t256 + 255) / 256, 256, 0, stream>>>(bufB, b2, t256, 8, 1);

  // ---- layer 3: out = Agg(h2 @ W3) + b3  (no relu) ----
  gemm_wmma_f32<128, false><<<MT, 256, 0, stream>>>(bufB, W3, bufA);
  agg_init_k     <<<(t128 + 255) / 256, 256, 0, stream>>>(bufA, dis, out, t128, 7);
  agg_edges_k<128><<<EB, 256, 0, stream>>>(bufA, dis, src, dst, out, N_EDGES);
  bias_act_k     <<<(t128 + 255) / 256, 256, 0, stream>>>(out, b3, t128, 7, 0);
}